// HashEncoder_62491774157121
// MI455X (gfx1250) — compile-verified
//
#include <hip/hip_runtime.h>
#include <math.h>

typedef __attribute__((ext_vector_type(2))) float v2f;
typedef __attribute__((ext_vector_type(8))) float v8f;

#define HPI1 2654435761u
#define HPI2 805459861u

struct LvlP {
  int n[8]; int k[8]; int m[8]; int lidx[8];
  long long goff[8];   // float offset of level grid in ws
  long long cs[9];     // cell prefix sums within this group
  int nlv;
};

struct Lvl2 { int m[16]; long long goff[16]; };

// ---------------- Phase 1a: wave-per-cell pooling (k >= 4) ----------------
__global__ __launch_bounds__(256)
void pool_wave(const float* __restrict__ x, const float* __restrict__ ht,
               float* __restrict__ g, LvlP P) {
  // wave id is uniform: pin it to an SGPR so decode/addressing go scalar
  int wid = __builtin_amdgcn_readfirstlane(
      (int)((blockIdx.x * blockDim.x + threadIdx.x) >> 5));
  int lane = threadIdx.x & 31;
  if ((long long)wid >= P.cs[P.nlv]) return;       // scalar branch
  int lv = 0;
  while (lv + 1 < P.nlv && (long long)wid >= P.cs[lv + 1]) ++lv;

  int m = P.m[lv], k = P.k[lv], n = P.n[lv];
  int local = (int)((long long)wid - P.cs[lv]);
  int m3 = m * m * m;
  int b  = local / m3;
  int rc = local - b * m3;
  int a  = rc / (m * m);
  int bb = (rc / m) % m;
  int cc = rc % m;

  int k3 = k * k * k;
  unsigned gi[3];
  for (int ch = 0; ch < 3; ++ch) {
    float mx = -1e30f;
    for (int t = lane; t < k3; t += 32) {
      int dh = t / (k * k);
      int rr = t - dh * k * k;
      int dw = rr / k;
      int dd = rr - dw * k;
      size_t idx = (((size_t)(b * 3 + ch) * 64 + (a * k + dh)) * 64 + (bb * k + dw)) * 64
                   + (cc * k + dd);
      mx = fmaxf(mx, x[idx]);
    }
    for (int off = 16; off > 0; off >>= 1)
      mx = fmaxf(mx, __shfl_xor(mx, off, 32));
    gi[ch] = (unsigned)(mx * (float)n);      // trunc toward zero (x >= 0)
  }
  unsigned h8 = (gi[0] ^ (gi[1] * HPI1) ^ (gi[2] * HPI2)) & 255u;
  if (lane < 3) {
    int L = P.lidx[lv];
    float feat = ht[((size_t)L * 256 + h8) * 3 + lane];
    // grid layout: [b][f][p=a][q=cc][r=bb]  (output h<-a, w<-cc(nd), d<-bb(nw))
    g[P.goff[lv] + (((size_t)(b * 3 + lane) * m + a) * m + cc) * m + bb] = feat;
  }
}

// ---------------- Phase 1b: thread-per-cell pooling (k <= 3) ----------------
__global__ __launch_bounds__(256)
void pool_thread(const float* __restrict__ x, const float* __restrict__ ht,
                 float* __restrict__ g, LvlP P) {
  long long tid = (long long)blockIdx.x * blockDim.x + threadIdx.x;
  if (tid >= P.cs[P.nlv]) return;
  int lv = 0;
  while (lv + 1 < P.nlv && tid >= P.cs[lv + 1]) ++lv;

  int m = P.m[lv], k = P.k[lv], n = P.n[lv];
  long long local = tid - P.cs[lv];
  int m3 = m * m * m;
  int b  = (int)(local / m3);
  int rc = (int)(local - (long long)b * m3);
  int a  = rc / (m * m);
  int bb = (rc / m) % m;
  int cc = rc % m;

  unsigned gi[3];
  for (int ch = 0; ch < 3; ++ch) {
    float mx = -1e30f;
    for (int dh = 0; dh < k; ++dh)
      for (int dw = 0; dw < k; ++dw)
        for (int dd = 0; dd < k; ++dd) {
          size_t idx = (((size_t)(b * 3 + ch) * 64 + (a * k + dh)) * 64 + (bb * k + dw)) * 64
                       + (cc * k + dd);
          mx = fmaxf(mx, x[idx]);
        }
    gi[ch] = (unsigned)(mx * (float)n);
  }
  unsigned h8 = (gi[0] ^ (gi[1] * HPI1) ^ (gi[2] * HPI2)) & 255u;
  int L = P.lidx[lv];
  for (int f = 0; f < 3; ++f) {
    float feat = ht[((size_t)L * 256 + h8) * 3 + f];
    g[P.goff[lv] + (((size_t)(b * 3 + f) * m + a) * m + cc) * m + bb] = feat;
  }
}

// ---------------- Phase 2: WMMA trilinear upsample ----------------
// Branchless footprint of jax linear resize (upsample) for one output coord:
// contributes w0 at index e0 and w1 at index e1 (e0==e1 with w0=1 at edges).
__device__ __forceinline__ void axis_foot(int o, int m, float scale,
                                          int& e0, int& e1, float& w0, float& w1) {
  float t = ((float)o + 0.5f) * scale - 0.5f;
  float fl = floorf(t);
  int j = (int)fl;
  float f = t - fl;
  bool lo = (j < 0);
  bool hi = (j >= m - 1);
  int jc = lo ? 0 : (hi ? (m - 1) : j);
  e0 = jc;
  e1 = (lo | hi) ? jc : (j + 1);
  w0 = (lo | hi) ? 1.0f : (1.0f - f);
  w1 = (lo | hi) ? 0.0f : f;
}

__device__ __forceinline__ float foot_wgt(int kk, int e0, int e1, float w0, float w1) {
  return (kk == e0) ? w0 : ((kk == e1) ? w1 : 0.0f);
}

__global__ __launch_bounds__(256)
void upsample_wmma(const float* __restrict__ g, float* __restrict__ out, Lvl2 P) {
  // wave id is uniform: pin to SGPR -> tile decode, m, nch, loop control and
  // the tail-chunk branch all become scalar; EXEC stays all-ones around WMMA.
  int wid = __builtin_amdgcn_readfirstlane(
      (int)((blockIdx.x * blockDim.x + threadIdx.x) >> 5));   // 49152 waves exactly
  int lane = threadIdx.x & 31;
  int l = wid / 3072;
  int r = wid - l * 3072;
  int b = r / 768;  r -= b * 768;
  int f = r / 256;  r -= f * 256;
  int h   = r >> 2;
  int w0c = (r & 3) << 4;

  int m = P.m[l];
  float scale = (float)m * (1.0f / 64.0f);
  const float* G = g + P.goff[l] + (size_t)(b * 3 + f) * (m * m * m);

  // h footprint (scalar), w footprint (per lane row)
  int p0, p1; float wh0, wh1; axis_foot(h, m, scale, p0, p1, wh0, wh1);
  int wl = w0c + (lane & 15);
  int q0, q1; float ww0, ww1; axis_foot(wl, m, scale, q0, q1, ww0, ww1);

  const float* r00 = G + ((size_t)p0 * m + q0) * m;
  const float* r01 = G + ((size_t)p0 * m + q1) * m;
  const float* r10 = G + ((size_t)p1 * m + q0) * m;
  const float* r11 = G + ((size_t)p1 * m + q1) * m;
  float c00 = wh0 * ww0;
  float c01 = wh0 * ww1;
  float c10 = wh1 * ww0;
  float c11 = wh1 * ww1;

  int kb   = (lane >> 4) << 1;   // A/B fragment K base within chunk: 0 or 2 (even)
  int dloc = lane & 15;          // B fragment N index

  // d-axis footprints: loop-invariant per lane, one per N-tile
  int e0a, e1a; float u0a, u1a; axis_foot( 0 + dloc, m, scale, e0a, e1a, u0a, u1a);
  int e0b, e1b; float u0b, u1b; axis_foot(16 + dloc, m, scale, e0b, e1b, u0b, u1b);
  int e0c, e1c; float u0c, u1c; axis_foot(32 + dloc, m, scale, e0c, e1c, u0c, u1c);
  int e0d, e1d; float u0d, u1d; axis_foot(48 + dloc, m, scale, e0d, e1d, u0d, u1d);

  v8f acc0 = {}; v8f acc1 = {}; v8f acc2 = {}; v8f acc3 = {};
  int nch = (m + 3) >> 2;
  for (int c = 0; c < nch; ++c) {
    int kkA = 4 * c + kb;                 // even -> 8B-aligned pair load
    v2f A;
    if (4 * c + 3 < m) {                  // scalar branch (c, m uniform in SGPRs)
      v2f a0 = *(const v2f*)(r00 + kkA);
      v2f a1 = *(const v2f*)(r01 + kkA);
      v2f a2 = *(const v2f*)(r10 + kkA);
      v2f a3 = *(const v2f*)(r11 + kkA);
      A = c00 * a0 + c01 * a1 + c10 * a2 + c11 * a3;
    } else {                              // tail chunk: clamped scalar taps
      int s0 = kkA     < m ? kkA     : m - 1;
      int s1 = kkA + 1 < m ? kkA + 1 : m - 1;
      A.x = c00 * r00[s0] + c01 * r01[s0] + c10 * r10[s0] + c11 * r11[s0];
      A.y = c00 * r00[s1] + c01 * r01[s1] + c10 * r10[s1] + c11 * r11[s1];
    }
    int k0 = kkA, k1 = kkA + 1;
    v2f B;
    B.x = foot_wgt(k0, e0a, e1a, u0a, u1a);
    B.y = foot_wgt(k1, e0a, e1a, u0a, u1a);
    acc0 = __builtin_amdgcn_wmma_f32_16x16x4_f32(false, A, false, B, (short)0, acc0, false, false);
    B.x = foot_wgt(k0, e0b, e1b, u0b, u1b);
    B.y = foot_wgt(k1, e0b, e1b, u0b, u1b);
    acc1 = __builtin_amdgcn_wmma_f32_16x16x4_f32(false, A, false, B, (short)0, acc1, false, false);
    B.x = foot_wgt(k0, e0c, e1c, u0c, u1c);
    B.y = foot_wgt(k1, e0c, e1c, u0c, u1c);
    acc2 = __builtin_amdgcn_wmma_f32_16x16x4_f32(false, A, false, B, (short)0, acc2, false, false);
    B.x = foot_wgt(k0, e0d, e1d, u0d, u1d);
    B.y = foot_wgt(k1, e0d, e1d, u0d, u1d);
    acc3 = __builtin_amdgcn_wmma_f32_16x16x4_f32(false, A, false, B, (short)0, acc3, false, false);
  }

  // D layout: VGPR v -> (M = v + (lane>=16 ? 8 : 0), N = lane&15)
  size_t obase = (((size_t)b * 48 + (size_t)(l * 3 + f)) * 64 + h) * 4096;
  int mb = w0c + ((lane >> 4) << 3);
#pragma unroll
  for (int v = 0; v < 8; ++v) {
    size_t row = obase + (size_t)(mb + v) * 64;
    __builtin_nontemporal_store(acc0[v], &out[row +  0 + dloc]);
    __builtin_nontemporal_store(acc1[v], &out[row + 16 + dloc]);
    __builtin_nontemporal_store(acc2[v], &out[row + 32 + dloc]);
    __builtin_nontemporal_store(acc3[v], &out[row + 48 + dloc]);
  }
}

extern "C" void kernel_launch(void* const* d_in, const int* in_sizes, int n_in,
                              void* d_out, int out_size, void* d_ws, size_t ws_size,
                              hipStream_t stream) {
  const float* x  = (const float*)d_in[0];
  const float* ht = (const float*)d_in[1];
  float* out = (float*)d_out;
  float* ws  = (float*)d_ws;

  // Per-level geometry (same double math as reference)
  int n[16], k[16], m[16];
  long long goff[16];
  double growth = exp(log(64.0 / 4.0) / 15.0);
  long long tot = 0;
  for (int l = 0; l < 16; ++l) {
    n[l] = (int)floor(4.0 * pow(growth, (double)l));
    k[l] = 64 / n[l];
    m[l] = 64 / k[l];
    goff[l] = tot;
    tot += 12LL * m[l] * m[l] * m[l];   // 4 batch * 3 features
  }
  (void)ws_size;  // needs ~55 MB; assumed provided

  // Split levels: k>=4 -> wave kernel, else thread kernel
  LvlP Pa; Pa.nlv = 0; Pa.cs[0] = 0;
  LvlP Pb; Pb.nlv = 0; Pb.cs[0] = 0;
  for (int l = 0; l < 16; ++l) {
    long long cells = 4LL * m[l] * m[l] * m[l];
    LvlP* P = (k[l] >= 4) ? &Pa : &Pb;
    int i = P->nlv;
    P->n[i] = n[l]; P->k[i] = k[l]; P->m[i] = m[l]; P->lidx[i] = l;
    P->goff[i] = goff[l];
    P->cs[i + 1] = P->cs[i] + cells;
    P->nlv = i + 1;
  }

  // Phase 1a: wave per cell
  {
    long long waves = Pa.cs[Pa.nlv];
    long long threads = waves * 32;
    int blocks = (int)((threads + 255) / 256);
    pool_wave<<<blocks, 256, 0, stream>>>(x, ht, ws, Pa);
  }
  // Phase 1b: thread per cell
  {
    long long threads = Pb.cs[Pb.nlv];
    int blocks = (int)((threads + 255) / 256);
    pool_thread<<<blocks, 256, 0, stream>>>(x, ht, ws, Pb);
  }
  // Phase 2: WMMA upsample. 16 lvl * 4 b * 3 f * 64 h * 4 w-chunks = 49152 waves
  {
    Lvl2 P2;
    for (int l = 0; l < 16; ++l) { P2.m[l] = m[l]; P2.goff[l] = goff[l]; }
    int blocks = 49152 / 8;   // 8 waves per 256-thread block, exact
    upsample_wmma<<<blocks, 256, 0, stream>>>(ws, out, P2);
  }
}